// MsaHmmLayer_79113297592749
// MI455X (gfx1250) — compile-verified
//
#include <hip/hip_runtime.h>
#include <hip/hip_bf16.h>

typedef float v2f __attribute__((ext_vector_type(2)));
typedef float v8f __attribute__((ext_vector_type(8)));

#define MM 2
#define BB 64
#define LL 512
#define SS 26
#define QQ 512

#define ROWS 16          // batch rows per workgroup (WMMA M)
#define A_STRIDE 516     // lds_alpha row stride in dwords; 516%64==4 -> 16 rows hit 16 banks
#define IN_STRIDE 36     // lds_in row stride in dwords;  36%64 cycles 16 distinct banks

// D = A(16x4) * B(4x16) + C, all f32, wave32.
// A: lane L holds row L&15, VGPR0=K(2*half), VGPR1=K(2*half+1)
// B: lane L holds col L&15, VGPR0=row K(2*half), VGPR1=row K(2*half+1)
// C/D: VGPR j holds row j (lanes 0-15) / j+8 (lanes 16-31), col = lane&15

// Sum across each 16-lane DPP row with VALU-only butterflies (no LDS, no waits):
// xor1 = quad_perm(1,0,3,2)=0xB1, xor2 = quad_perm(2,3,0,1)=0x4E,
// cross-quad = row_half_mirror=0x141, cross-8 = row_mirror=0x140.
__device__ __forceinline__ float half16_sum(float v) {
    v += __int_as_float(__builtin_amdgcn_update_dpp(0, __float_as_int(v), 0xB1,  0xF, 0xF, true));
    v += __int_as_float(__builtin_amdgcn_update_dpp(0, __float_as_int(v), 0x4E,  0xF, 0xF, true));
    v += __int_as_float(__builtin_amdgcn_update_dpp(0, __float_as_int(v), 0x141, 0xF, 0xF, true));
    v += __int_as_float(__builtin_amdgcn_update_dpp(0, __float_as_int(v), 0x140, 0xF, 0xF, true));
    return v;
}

__global__ __launch_bounds__(1024)
void hmm_forward_kernel(const float* __restrict__ inputs,
                        const float* __restrict__ init_dist,
                        const float* __restrict__ trans,
                        const float* __restrict__ emit,
                        float* __restrict__ out)
{
    __shared__ float lds_alpha[ROWS * A_STRIDE];   // 33,024 B
    __shared__ float lds_in[ROWS * IN_STRIDE];     //  2,304 B
    __shared__ float lds_rowsum[ROWS];

    const int tid  = threadIdx.x;
    const int lane = tid & 31;
    const int wave = tid >> 5;
    const int n    = lane & 15;           // col within tile (B/C/D) or A row
    const int half = lane >> 4;           // selects K pair {2h, 2h+1}
    const int q    = wave * 16 + n;       // global state index (N)

    const int m  = blockIdx.x >> 2;       // model
    const int rb = blockIdx.x & 3;        // 16-row batch block

    const float* in_m    = inputs + (size_t)(m * BB + rb * ROWS) * LL * SS;
    const float* trans_m = trans  + (size_t)m * QQ * QQ;
    const float* emit_m  = emit   + (size_t)m * SS * QQ;
    float* out_fwd = out + (size_t)(m * BB + rb * ROWS) * LL * QQ;
    float* out_ll  = out + (size_t)MM * BB * LL * QQ;

    const float initv = init_dist[m * QQ + q];
    const int   kaoff = 2 * half;
    // per-lane base for trans B loads: trans[(k+2h)*Q + q], second row at +Q
    const float* tb = trans_m + (size_t)kaoff * QQ + q;

    // Emission B operand is t-invariant: preload 7 K-pair slices into registers
    // (zero-padded past S=26).
    v2f ebv[7];
#pragma unroll
    for (int i = 0; i < 7; ++i) {
        const int kA = 4 * i + kaoff;
        ebv[i].x = (kA     < SS) ? emit_m[(size_t)kA * QQ + q]       : 0.0f;
        ebv[i].y = (kA + 1 < SS) ? emit_m[((size_t)kA + 1) * QQ + q] : 0.0f;
    }

    float ll = 0.0f;  // meaningful in tid<16

    // ================= t = 0 (peeled) =================
    if (tid < ROWS * 32) {
        int r = tid >> 5, s = tid & 31;
        lds_in[r * IN_STRIDE + s] = (s < SS) ? in_m[(size_t)r * LL * SS + s] : 0.0f;
    }
    if (tid < ROWS) lds_rowsum[tid] = 0.0f;
    __syncthreads();

    {
        v8f e = {};
#pragma unroll
        for (int i = 0; i < 7; ++i) {
            v2f a = *(const v2f*)&lds_in[n * IN_STRIDE + 4 * i + kaoff];
            e = __builtin_amdgcn_wmma_f32_16x16x4_f32(false, a, false, ebv[i],
                                                      (short)0, e, false, false);
        }
        v8f a_t;
#pragma unroll
        for (int j = 0; j < 8; ++j) a_t[j] = e[j] * initv;   // init_dist[q] * e0

        float p[8];
#pragma unroll
        for (int j = 0; j < 8; ++j) p[j] = half16_sum(a_t[j]);
        if (n == 0) {   // one exec toggle for all 8 cross-wave atomics
#pragma unroll
            for (int j = 0; j < 8; ++j)
                atomicAdd(&lds_rowsum[j + 8 * half], p[j]);  // ds_add_f32
        }
        __syncthreads();
#pragma unroll
        for (int j = 0; j < 8; ++j) {
            const int row = j + 8 * half;
            const float al = a_t[j] * __builtin_amdgcn_rcpf(lds_rowsum[row]);
            lds_alpha[row * A_STRIDE + q] = al;
            out_fwd[(size_t)row * LL * QQ + q] = al;
        }
        if (tid < ROWS) ll += __logf(lds_rowsum[tid]);
    }

    // ================= t = 1 .. 511 =================
#pragma unroll 1
    for (int t = 1; t < LL; ++t) {
        __syncthreads();  // prior step's rowsum reads + alpha writes complete
        if (tid < ROWS * 32) {
            int r = tid >> 5, s = tid & 31;
            lds_in[r * IN_STRIDE + s] =
                (s < SS) ? in_m[((size_t)r * LL + t) * SS + s] : 0.0f;
        }
        if (tid < ROWS) lds_rowsum[tid] = 0.0f;
        __syncthreads();

        // ---- emission tile: (16 x 26) @ (26 x 16), K zero-padded to 28 ----
        v8f e = {};
#pragma unroll
        for (int i = 0; i < 7; ++i) {
            v2f a = *(const v2f*)&lds_in[n * IN_STRIDE + 4 * i + kaoff];
            e = __builtin_amdgcn_wmma_f32_16x16x4_f32(false, a, false, ebv[i],
                                                      (short)0, e, false, false);
        }

        // ---- (16 x 512) @ (512 x 16): 128 WMMAs, 4 independent acc chains ----
        v8f acc0 = {}, acc1 = {}, acc2 = {}, acc3 = {};
#pragma unroll 2
        for (int k = 0; k < QQ; k += 16) {
            {
                v2f a = *(const v2f*)&lds_alpha[n * A_STRIDE + k + kaoff];
                v2f b; b.x = tb[(size_t)k * QQ]; b.y = tb[(size_t)k * QQ + QQ];
                acc0 = __builtin_amdgcn_wmma_f32_16x16x4_f32(false, a, false, b, (short)0, acc0, false, false);
            }
            {
                v2f a = *(const v2f*)&lds_alpha[n * A_STRIDE + (k + 4) + kaoff];
                v2f b; b.x = tb[(size_t)(k + 4) * QQ]; b.y = tb[(size_t)(k + 4) * QQ + QQ];
                acc1 = __builtin_amdgcn_wmma_f32_16x16x4_f32(false, a, false, b, (short)0, acc1, false, false);
            }
            {
                v2f a = *(const v2f*)&lds_alpha[n * A_STRIDE + (k + 8) + kaoff];
                v2f b; b.x = tb[(size_t)(k + 8) * QQ]; b.y = tb[(size_t)(k + 8) * QQ + QQ];
                acc2 = __builtin_amdgcn_wmma_f32_16x16x4_f32(false, a, false, b, (short)0, acc2, false, false);
            }
            {
                v2f a = *(const v2f*)&lds_alpha[n * A_STRIDE + (k + 12) + kaoff];
                v2f b; b.x = tb[(size_t)(k + 12) * QQ]; b.y = tb[(size_t)(k + 12) * QQ + QQ];
                acc3 = __builtin_amdgcn_wmma_f32_16x16x4_f32(false, a, false, b, (short)0, acc3, false, false);
            }
        }
        v8f s = (acc0 + acc1) + (acc2 + acc3);
        v8f a_t;
#pragma unroll
        for (int j = 0; j < 8; ++j) a_t[j] = s[j] * e[j];

        // ---- row sums: DPP butterflies per half, then one guarded atomic run ----
        float p[8];
#pragma unroll
        for (int j = 0; j < 8; ++j) p[j] = half16_sum(a_t[j]);
        if (n == 0) {   // lanes 0 and 16: single saveexec, 8 back-to-back ds_add_f32
#pragma unroll
            for (int j = 0; j < 8; ++j)
                atomicAdd(&lds_rowsum[j + 8 * half], p[j]);
        }
        __syncthreads();

        // ---- normalize (v_rcp), stash alpha in LDS, write forward ----
#pragma unroll
        for (int j = 0; j < 8; ++j) {
            const int row = j + 8 * half;
            const float al = a_t[j] * __builtin_amdgcn_rcpf(lds_rowsum[row]);
            lds_alpha[row * A_STRIDE + q] = al;
            out_fwd[((size_t)row * LL + t) * QQ + q] = al;   // coalesced 64B per half
        }
        if (tid < ROWS) ll += __logf(lds_rowsum[tid]);

        if (t + 1 < LL && tid < ROWS)
            __builtin_prefetch(&in_m[((size_t)tid * LL + (t + 1)) * SS], 0, 1);
    }

    if (tid < ROWS) out_ll[m * BB + rb * ROWS + tid] = ll;
}

extern "C" void kernel_launch(void* const* d_in, const int* in_sizes, int n_in,
                              void* d_out, int out_size, void* d_ws, size_t ws_size,
                              hipStream_t stream) {
    (void)in_sizes; (void)n_in; (void)out_size; (void)d_ws; (void)ws_size;
    const float* inputs    = (const float*)d_in[0];  // (M,B,L,S)
    const float* init_dist = (const float*)d_in[1];  // (M,Q)
    const float* trans     = (const float*)d_in[2];  // (M,Q,Q)
    const float* emit      = (const float*)d_in[3];  // (M,S,Q)
    float* out = (float*)d_out;                      // forward (M,B,L,Q) ++ loglik (M,B)

    dim3 grid(MM * (BB / ROWS));   // 8 workgroups
    dim3 block(1024);              // 32 waves, one 16-col tile each
    hmm_forward_kernel<<<grid, block, 0, stream>>>(inputs, init_dist, trans, emit, out);
}